// GNCA_62826781606046
// MI455X (gfx1250) — compile-verified
//
#include <hip/hip_runtime.h>
#include <hip/hip_bf16.h>
#include <cstddef>

#define N_NODES 100000
#define N_EDGES 600000
#define D 128
#define H 256

typedef float v2f __attribute__((ext_vector_type(2)));
typedef float v8f __attribute__((ext_vector_type(8)));

__device__ __forceinline__ v8f wmma_f32(v2f a, v2f b, v8f c) {
    // D = A(16x4 f32) * B(4x16 f32) + C(16x16 f32)
    // (neg_a, A, neg_b, B, c_mod, C, reuse_a, reuse_b)
    return __builtin_amdgcn_wmma_f32_16x16x4_f32(false, a, false, b, (short)0, c,
                                                 false, false);
}

// Tile geometry: 32 rows per block, 8 waves per block.
#define ROWS 32
#define XS_STRIDE 132   // 128 + 4 pad -> conflict-free A-fragment reads
#define H1_STRIDE 260   // 256 + 4 pad

// --------------------------------------------------------------------------
// Kernel 1: message MLP  m = relu(X @ Wm1 + bm1) @ Wm2 + bm2
// Each wave owns col-tiles and carries TWO row accumulators (rows 0-15 and
// 16-31) so every B (weight) fragment load feeds 2 WMMAs.
// --------------------------------------------------------------------------
__global__ __launch_bounds__(256) void gnn_msg_mlp(
    const float* __restrict__ nodes, const float* __restrict__ Wm1,
    const float* __restrict__ bm1, const float* __restrict__ Wm2,
    const float* __restrict__ bm2, float* __restrict__ m_out) {
    __shared__ float Xs[ROWS * XS_STRIDE];    // 16896 B
    __shared__ float H1s[ROWS * H1_STRIDE];   // 33280 B

    const int tid = threadIdx.x;
    const int row_base = blockIdx.x * ROWS;

    // Cooperative load of the 32x128 node tile (float4, zero-pad tail rows).
    for (int i = tid; i < ROWS * (D / 4); i += 256) {
        int row = i / (D / 4);
        int c4 = i % (D / 4);
        int rg = row_base + row;
        float4 v = make_float4(0.f, 0.f, 0.f, 0.f);
        if (rg < N_NODES)
            v = ((const float4*)(nodes + (size_t)rg * D))[c4];
        *(float4*)&Xs[row * XS_STRIDE + c4 * 4] = v;
    }
    __syncthreads();

    const int w = tid >> 5;
    const int lane = tid & 31;
    const int half = lane >> 4;   // 0: K=0,1   1: K=2,3
    const int l16 = lane & 15;

    // ---- Layer 1: [32x128] @ [128x256] -> relu -> H1s ----
    // 16 col-tiles / 8 waves = 2 per wave; 2 row accumulators each.
    for (int c = 0; c < 2; ++c) {
        int col = (w * 2 + c) * 16 + l16;
        float bv = bm1[col];
        v8f acc0 = {bv, bv, bv, bv, bv, bv, bv, bv};
        v8f acc1 = acc0;
#pragma unroll 4
        for (int k = 0; k < D; k += 4) {
            int kk = k + half * 2;
            v2f b = {Wm1[kk * H + col], Wm1[(kk + 1) * H + col]};
            const float* x0 = &Xs[l16 * XS_STRIDE + kk];
            const float* x1 = &Xs[(16 + l16) * XS_STRIDE + kk];
            v2f a0 = {x0[0], x0[1]};
            v2f a1 = {x1[0], x1[1]};
            acc0 = wmma_f32(a0, b, acc0);
            acc1 = wmma_f32(a1, b, acc1);
        }
#pragma unroll
        for (int r = 0; r < 8; ++r) {
            int row = r + half * 8;
            float v0 = acc0[r];
            float v1 = acc1[r];
            H1s[row * H1_STRIDE + col] = v0 > 0.f ? v0 : 0.f;
            H1s[(16 + row) * H1_STRIDE + col] = v1 > 0.f ? v1 : 0.f;
        }
    }
    __syncthreads();

    // ---- Layer 2: [32x256] @ [256x128] + bm2 -> m_out ----
    // 8 col-tiles / 8 waves = 1 per wave; 2 row accumulators.
    {
        int col = w * 16 + l16;
        float bv = bm2[col];
        v8f acc0 = {bv, bv, bv, bv, bv, bv, bv, bv};
        v8f acc1 = acc0;
#pragma unroll 4
        for (int k = 0; k < H; k += 4) {
            int kk = k + half * 2;
            v2f b = {Wm2[kk * D + col], Wm2[(kk + 1) * D + col]};
            const float* h0 = &H1s[l16 * H1_STRIDE + kk];
            const float* h1 = &H1s[(16 + l16) * H1_STRIDE + kk];
            v2f a0 = {h0[0], h0[1]};
            v2f a1 = {h1[0], h1[1]};
            acc0 = wmma_f32(a0, b, acc0);
            acc1 = wmma_f32(a1, b, acc1);
        }
#pragma unroll
        for (int r = 0; r < 8; ++r) {
            int row = r + half * 8;
            int rg0 = row_base + row;
            int rg1 = row_base + 16 + row;
            if (rg0 < N_NODES) m_out[(size_t)rg0 * D + col] = acc0[r];
            if (rg1 < N_NODES) m_out[(size_t)rg1 * D + col] = acc1[r];
        }
    }
}

// --------------------------------------------------------------------------
// Kernel 2: edge scatter  agg[recv[e]] += m[send[e]]   (one wave per edge)
// --------------------------------------------------------------------------
__global__ __launch_bounds__(256) void gnn_scatter(
    const float* __restrict__ m, const int* __restrict__ senders,
    const int* __restrict__ receivers, float* __restrict__ agg) {
    const int wave = threadIdx.x >> 5;
    const int lane = threadIdx.x & 31;
    const int e = blockIdx.x * 8 + wave;
    if (e >= N_EDGES) return;
    const int s = senders[e];
    const int r = receivers[e];
    const float4 v = ((const float4*)(m + (size_t)s * D))[lane];
    float* dst = agg + (size_t)r * D + lane * 4;
    __hip_atomic_fetch_add(dst + 0, v.x, __ATOMIC_RELAXED, __HIP_MEMORY_SCOPE_AGENT);
    __hip_atomic_fetch_add(dst + 1, v.y, __ATOMIC_RELAXED, __HIP_MEMORY_SCOPE_AGENT);
    __hip_atomic_fetch_add(dst + 2, v.z, __ATOMIC_RELAXED, __HIP_MEMORY_SCOPE_AGENT);
    __hip_atomic_fetch_add(dst + 3, v.w, __ATOMIC_RELAXED, __HIP_MEMORY_SCOPE_AGENT);
}

// --------------------------------------------------------------------------
// Kernel 3: node MLP  out = X + relu([X||agg] @ Wn1 + bn1) @ Wn2 + bn2
// agg aliases `out`: every agg read happens before the barrier; each block
// overwrites only its own rows afterwards.
// --------------------------------------------------------------------------
__global__ __launch_bounds__(256) void gnn_node_mlp(
    const float* __restrict__ nodes, const float* __restrict__ agg,
    const float* __restrict__ Wn1, const float* __restrict__ bn1,
    const float* __restrict__ Wn2, const float* __restrict__ bn2,
    float* __restrict__ out) {
    __shared__ float Xs[ROWS * XS_STRIDE];
    __shared__ float H1s[ROWS * H1_STRIDE];

    const int tid = threadIdx.x;
    const int row_base = blockIdx.x * ROWS;

    for (int i = tid; i < ROWS * (D / 4); i += 256) {
        int row = i / (D / 4);
        int c4 = i % (D / 4);
        int rg = row_base + row;
        float4 v = make_float4(0.f, 0.f, 0.f, 0.f);
        if (rg < N_NODES)
            v = ((const float4*)(nodes + (size_t)rg * D))[c4];
        *(float4*)&Xs[row * XS_STRIDE + c4 * 4] = v;
    }
    __syncthreads();

    const int w = tid >> 5;
    const int lane = tid & 31;
    const int half = lane >> 4;
    const int l16 = lane & 15;

    // Per-lane agg row pointers (clamped; garbage rows never stored).
    int rg0 = row_base + l16;
    int rg1 = row_base + 16 + l16;
    if (rg0 >= N_NODES) rg0 = N_NODES - 1;
    if (rg1 >= N_NODES) rg1 = N_NODES - 1;
    const float* aggRow0 = agg + (size_t)rg0 * D;
    const float* aggRow1 = agg + (size_t)rg1 * D;

    // ---- Layer 1: [32 x 256(=X||agg)] @ [256x256] -> relu -> H1s ----
    for (int c = 0; c < 2; ++c) {
        int col = (w * 2 + c) * 16 + l16;
        float bv = bn1[col];
        v8f acc0 = {bv, bv, bv, bv, bv, bv, bv, bv};
        v8f acc1 = acc0;
#pragma unroll 4
        for (int k = 0; k < D; k += 4) {  // X half of the concat
            int kk = k + half * 2;
            v2f b = {Wn1[kk * H + col], Wn1[(kk + 1) * H + col]};
            const float* x0 = &Xs[l16 * XS_STRIDE + kk];
            const float* x1 = &Xs[(16 + l16) * XS_STRIDE + kk];
            v2f a0 = {x0[0], x0[1]};
            v2f a1 = {x1[0], x1[1]};
            acc0 = wmma_f32(a0, b, acc0);
            acc1 = wmma_f32(a1, b, acc1);
        }
#pragma unroll 4
        for (int k = D; k < 2 * D; k += 4) {  // agg half of the concat
            int kk = k + half * 2;
            v2f b = {Wn1[kk * H + col], Wn1[(kk + 1) * H + col]};
            const float* a0p = aggRow0 + (kk - D);
            const float* a1p = aggRow1 + (kk - D);
            v2f a0 = {a0p[0], a0p[1]};
            v2f a1 = {a1p[0], a1p[1]};
            acc0 = wmma_f32(a0, b, acc0);
            acc1 = wmma_f32(a1, b, acc1);
        }
#pragma unroll
        for (int r = 0; r < 8; ++r) {
            int row = r + half * 8;
            float v0 = acc0[r];
            float v1 = acc1[r];
            H1s[row * H1_STRIDE + col] = v0 > 0.f ? v0 : 0.f;
            H1s[(16 + row) * H1_STRIDE + col] = v1 > 0.f ? v1 : 0.f;
        }
    }
    __syncthreads();

    // ---- Layer 2: [32x256] @ [256x128] + bn2 + residual -> out ----
    {
        int col = w * 16 + l16;
        float bv = bn2[col];
        v8f acc0 = {bv, bv, bv, bv, bv, bv, bv, bv};
        v8f acc1 = acc0;
#pragma unroll 4
        for (int k = 0; k < H; k += 4) {
            int kk = k + half * 2;
            v2f b = {Wn2[kk * D + col], Wn2[(kk + 1) * D + col]};
            const float* h0 = &H1s[l16 * H1_STRIDE + kk];
            const float* h1 = &H1s[(16 + l16) * H1_STRIDE + kk];
            v2f a0 = {h0[0], h0[1]};
            v2f a1 = {h1[0], h1[1]};
            acc0 = wmma_f32(a0, b, acc0);
            acc1 = wmma_f32(a1, b, acc1);
        }
#pragma unroll
        for (int r = 0; r < 8; ++r) {
            int row = r + half * 8;
            int g0 = row_base + row;
            int g1 = row_base + 16 + row;
            if (g0 < N_NODES)
                out[(size_t)g0 * D + col] = Xs[row * XS_STRIDE + col] + acc0[r];
            if (g1 < N_NODES)
                out[(size_t)g1 * D + col] = Xs[(16 + row) * XS_STRIDE + col] + acc1[r];
        }
    }
}

// --------------------------------------------------------------------------
extern "C" void kernel_launch(void* const* d_in, const int* in_sizes, int n_in,
                              void* d_out, int out_size, void* d_ws, size_t ws_size,
                              hipStream_t stream) {
    (void)in_sizes; (void)n_in; (void)out_size; (void)ws_size;
    const float* nodes     = (const float*)d_in[0];
    const int*   senders   = (const int*)d_in[1];
    const int*   receivers = (const int*)d_in[2];
    const float* Wm1 = (const float*)d_in[3];
    const float* bm1 = (const float*)d_in[4];
    const float* Wm2 = (const float*)d_in[5];
    const float* bm2 = (const float*)d_in[6];
    const float* Wn1 = (const float*)d_in[7];
    const float* bn1 = (const float*)d_in[8];
    const float* Wn2 = (const float*)d_in[9];
    const float* bn2 = (const float*)d_in[10];

    float* m_buf = (float*)d_ws;           // [N_NODES, D] messages
    float* out   = (float*)d_out;          // doubles as agg buffer

    const int row_blocks = (N_NODES + ROWS - 1) / ROWS;  // 3125

    gnn_msg_mlp<<<row_blocks, 256, 0, stream>>>(nodes, Wm1, bm1, Wm2, bm2, m_buf);

    hipMemsetAsync(d_out, 0, sizeof(float) * (size_t)N_NODES * D, stream);

    gnn_scatter<<<(N_EDGES + 7) / 8, 256, 0, stream>>>(m_buf, senders, receivers, out);

    gnn_node_mlp<<<row_blocks, 256, 0, stream>>>(nodes, out, Wn1, bn1, Wn2, bn2, out);
}